// ResidualVectorQuantizer_35055523070490
// MI455X (gfx1250) — compile-verified
//
#include <hip/hip_runtime.h>

typedef float v2f __attribute__((ext_vector_type(2)));
typedef float v4f __attribute__((ext_vector_type(4)));
typedef float v8f __attribute__((ext_vector_type(8)));

#define N_CODES 256
#define E_DIM   128
#define NUM_Q   4
#define CB_STRIDE 132          // padded LDS row stride (floats); 132%64==4 -> conflict-free b64 loads
#define ROWS_PER_WAVE 16
#define WAVES_PER_BLOCK 8
#define ROWS_PER_BLOCK (ROWS_PER_WAVE*WAVES_PER_BLOCK)

// Residual VQ: scores via V_WMMA_F32_16X16X4_F32, codebook staged in CDNA5's 320KB LDS.
// Two n-tiles are processed per iteration with independent accumulators so the XDL
// pipe sees two interleavable WMMA chains instead of one serial D->C chain.
__global__ __launch_bounds__(256)
void rvq_wmma_kernel(const float* __restrict__ x,
                     const float* __restrict__ codebooks,   // [4,256,128]
                     float* __restrict__ xq,                // [N,128]
                     float* __restrict__ loss_sse,          // [4] accumulators in d_ws
                     float* __restrict__ idx_out,           // [N,4] written as float
                     int N)
{
    __shared__ float s_cb[N_CODES * CB_STRIDE];             // 132 KB padded codebook
    __shared__ float s_hn[N_CODES];                         // 0.5*||c||^2
    __shared__ int   s_idx[WAVES_PER_BLOCK][ROWS_PER_WAVE];

    const int tid  = threadIdx.x;
    const int wave = tid >> 5;
    const int lane = tid & 31;
    const int lm   = lane & 15;    // lane within half-wave
    const int hi   = lane >> 4;    // 0 -> K%4 in {0,1}; 1 -> K%4 in {2,3}  (fp32 A-frag layout)
    const int row0 = blockIdx.x * ROWS_PER_BLOCK + wave * ROWS_PER_WAVE;
    const int myrow = row0 + lm;   // lane pair (lm, lm+16) co-owns row lm of the tile

    // ---- residual = x, held in registers in WMMA A-fragment layout (64 VGPRs) ----
    float r[2 * (E_DIM / 4)];
    {
        const float* xrow = x + (size_t)myrow * E_DIM + 2 * hi;
        #pragma unroll
        for (int k = 0; k < E_DIM / 4; ++k) {
            v2f v = *(const v2f*)(xrow + 4 * k);
            r[2 * k]     = v.x;
            r[2 * k + 1] = v.y;
        }
    }

    for (int qi = 0; qi < NUM_Q; ++qi) {
        __syncthreads();   // protect s_cb from previous stage's readers
        // ---- stage codebook into padded LDS: thread i copies code row i ----
        {
            const float* src = codebooks + ((size_t)qi * N_CODES + tid) * E_DIM;
            #pragma unroll
            for (int j = 0; j < E_DIM / 4; ++j) {
                v4f v = *(const v4f*)(src + 4 * j);
                *(v4f*)&s_cb[tid * CB_STRIDE + 4 * j] = v;
            }
        }
        __syncthreads();
        {   // half squared norms
            float s = 0.f;
            #pragma unroll 8
            for (int j = 0; j < E_DIM; ++j) { float c = s_cb[tid * CB_STRIDE + j]; s += c * c; }
            s_hn[tid] = 0.5f * s;
        }
        __syncthreads();

        // ---- score(row,code) = r.c - 0.5||c||^2 ; argmin dist == argmax score ----
        float best[8];
        int   bidx[8];
        #pragma unroll
        for (int v = 0; v < 8; ++v) { best[v] = -3.4e38f; bidx[v] = 0; }

        for (int t = 0; t < N_CODES / 16; t += 2) {
            const int col0 = t * 16 + lm;                  // this lane's N column, tile t
            const int col1 = col0 + 16;                    // tile t+1
            const float hn0 = s_hn[col0];
            const float hn1 = s_hn[col1];
            v8f acc0, acc1;
            #pragma unroll
            for (int v = 0; v < 8; ++v) { acc0[v] = -hn0; acc1[v] = -hn1; }
            const float* bcol0 = &s_cb[col0 * CB_STRIDE + 2 * hi];
            const float* bcol1 = &s_cb[col1 * CB_STRIDE + 2 * hi];
            #pragma unroll
            for (int k = 0; k < E_DIM / 4; ++k) {          // 2x32 WMMAs, two independent chains
                v2f a; a.x = r[2 * k]; a.y = r[2 * k + 1]; // shared A frag (residual regs)
                v2f b0 = *(const v2f*)(bcol0 + 4 * k);
                v2f b1 = *(const v2f*)(bcol1 + 4 * k);
                acc0 = __builtin_amdgcn_wmma_f32_16x16x4_f32(
                        false, a, false, b0, (short)0, acc0, false, false);
                acc1 = __builtin_amdgcn_wmma_f32_16x16x4_f32(
                        false, a, false, b1, (short)0, acc1, false, false);
            }
            #pragma unroll
            for (int v = 0; v < 8; ++v) {                  // acc[v] = score(row v+8*hi, col)
                if (acc0[v] > best[v]) { best[v] = acc0[v]; bidx[v] = col0; }
                if (acc1[v] > best[v]) { best[v] = acc1[v]; bidx[v] = col1; }
            }
        }

        // ---- per-row argmax across the 16 lanes of each half-wave (wave32 shuffles) ----
        #pragma unroll
        for (int v = 0; v < 8; ++v) {
            float bv = best[v]; int bi = bidx[v];
            #pragma unroll
            for (int off = 8; off >= 1; off >>= 1) {
                float ov = __shfl_xor(bv, off, 32);
                int   oi = __shfl_xor(bi, off, 32);
                if (ov > bv || (ov == bv && oi < bi)) { bv = ov; bi = oi; }
            }
            if (lm == 0) {
                const int rrow = v + 8 * hi;
                s_idx[wave][rrow] = bi;
                idx_out[(size_t)(row0 + rrow) * NUM_Q + qi] = (float)bi;
            }
        }
        __syncthreads();

        // ---- residual -= quant ; accumulate SSE for the loss ----
        {
            const int ci = s_idx[wave][lm];
            const float* qrow = &s_cb[ci * CB_STRIDE + 2 * hi];
            float sse = 0.f;
            #pragma unroll
            for (int k = 0; k < E_DIM / 4; ++k) {
                v2f q = *(const v2f*)(qrow + 4 * k);
                float d0 = r[2 * k]     - q.x;
                float d1 = r[2 * k + 1] - q.y;
                r[2 * k] = d0; r[2 * k + 1] = d1;
                sse += d0 * d0 + d1 * d1;
            }
            #pragma unroll
            for (int off = 16; off >= 1; off >>= 1)
                sse += __shfl_xor(sse, off, 32);
            if (lane == 0) atomicAdd(&loss_sse[qi], sse);
        }
    }

    // ---- x_q = x - residual_final (x tile still hot in 192MB L2) ----
    {
        const float* xrow = x  + (size_t)myrow * E_DIM + 2 * hi;
        float*       orow = xq + (size_t)myrow * E_DIM + 2 * hi;
        #pragma unroll
        for (int k = 0; k < E_DIM / 4; ++k) {
            v2f v = *(const v2f*)(xrow + 4 * k);
            v2f o; o.x = v.x - r[2 * k]; o.y = v.y - r[2 * k + 1];
            *(v2f*)(orow + 4 * k) = o;
        }
    }
}

__global__ void rvq_finalize(const float* __restrict__ loss_sse,
                             float* __restrict__ loss_out, int N)
{
    // per stage: codebook_loss + BETA*commit_loss = 2 * SSE / (N*E_DIM); mean over stages
    float acc = 0.f;
    for (int qi = 0; qi < NUM_Q; ++qi)
        acc += 2.0f * loss_sse[qi] / ((float)N * (float)E_DIM);
    *loss_out = acc / (float)NUM_Q;
}

extern "C" void kernel_launch(void* const* d_in, const int* in_sizes, int n_in,
                              void* d_out, int out_size, void* d_ws, size_t ws_size,
                              hipStream_t stream)
{
    const float* x  = (const float*)d_in[0];
    const float* cb = (const float*)d_in[1];
    const int N = in_sizes[0] / E_DIM;                 // 262144

    float* out     = (float*)d_out;
    float* xq      = out;                              // [N*128]
    float* loss    = out + (size_t)N * E_DIM;          // [1]
    float* idx_out = loss + 1;                         // [N*4] as float
    float* sse     = (float*)d_ws;                     // [4] stage accumulators

    hipMemsetAsync(sse, 0, NUM_Q * sizeof(float), stream);

    dim3 block(256);
    dim3 grid(N / ROWS_PER_BLOCK);
    rvq_wmma_kernel<<<grid, block, 0, stream>>>(x, cb, xq, sse, idx_out, N);
    rvq_finalize<<<1, 1, 0, stream>>>(sse, loss, N);
}